// GCN_test_90993177133180
// MI455X (gfx1250) — compile-verified
//
#include <hip/hip_runtime.h>

#define NFEAT  128
#define NHID   64
#define NCLASS 16

typedef float v2f __attribute__((ext_vector_type(2)));
typedef float v8f __attribute__((ext_vector_type(8)));

// ---------------------------------------------------------------------------
// GEMM1: h_pre[N,64] = x[N,128] @ W1[128,64]   (f32 WMMA 16x16x4)
// One wave computes a 16-row x 64-col stripe (4 column tiles of 16).
// A layout (16x4 f32): lane l: m=l&15, h=l>>4; vgpr v holds A[m][k + v + 2h]
// B layout (4x16 f32): lane l: n=l&15, h=l>>4; vgpr v holds B[k + v + 2h][n]
// C/D layout: lane l: n=l&15; vgpr v holds D[v + 8*(l>>4)][n]
// ---------------------------------------------------------------------------
__global__ void gcn_gemm1(const float* __restrict__ x,
                          const float* __restrict__ W1,
                          float* __restrict__ h_pre,
                          int nTiles, int N) {
  const int wave = threadIdx.x >> 5;
  const int lane = threadIdx.x & 31;
  const int tile = blockIdx.x * 8 + wave;
  if (tile >= nTiles) return;                 // wave-uniform: EXEC stays all-1s

  const int r0 = tile * 16;
  const int m  = lane & 15;
  const int h  = lane >> 4;
  const int n  = lane & 15;

  int rowA = r0 + m;
  if (rowA >= N) rowA = N - 1;                // clamp loads, keep EXEC full
  const float* aPtr  = x  + (size_t)rowA * NFEAT + 2 * h;   // 8B aligned
  const float* bBase = W1 + (size_t)(2 * h) * NHID + n;

  v8f acc0 = {}, acc1 = {}, acc2 = {}, acc3 = {};

#pragma unroll 4
  for (int k = 0; k < NFEAT; k += 4) {
    v2f a = *(const v2f*)(aPtr + k);          // A[m][k+2h], A[m][k+2h+1]
    const float* bk = bBase + (size_t)k * NHID;
    v2f b0, b1, b2, b3;
    b0.x = bk[0];  b0.y = bk[NHID];
    b1.x = bk[16]; b1.y = bk[NHID + 16];
    b2.x = bk[32]; b2.y = bk[NHID + 32];
    b3.x = bk[48]; b3.y = bk[NHID + 48];
    acc0 = __builtin_amdgcn_wmma_f32_16x16x4_f32(false, a, false, b0, (short)0, acc0, false, false);
    acc1 = __builtin_amdgcn_wmma_f32_16x16x4_f32(false, a, false, b1, (short)0, acc1, false, false);
    acc2 = __builtin_amdgcn_wmma_f32_16x16x4_f32(false, a, false, b2, (short)0, acc2, false, false);
    acc3 = __builtin_amdgcn_wmma_f32_16x16x4_f32(false, a, false, b3, (short)0, acc3, false, false);
  }

#pragma unroll
  for (int v = 0; v < 8; ++v) {
    const int row = r0 + v + 8 * h;
    if (row < N) {
      float* o = h_pre + (size_t)row * NHID + n;
      o[0]  = acc0[v];
      o[16] = acc1[v];
      o[32] = acc2[v];
      o[48] = acc3[v];
    }
  }
}

// ---------------------------------------------------------------------------
// GEMM2: g[N,16] = relu(agg1[N,64]) @ W2[64,16]   (relu fused into A load)
// ---------------------------------------------------------------------------
__global__ void gcn_gemm2(const float* __restrict__ agg1,
                          const float* __restrict__ W2,
                          float* __restrict__ g,
                          int nTiles, int N) {
  const int wave = threadIdx.x >> 5;
  const int lane = threadIdx.x & 31;
  const int tile = blockIdx.x * 8 + wave;
  if (tile >= nTiles) return;

  const int r0 = tile * 16;
  const int m  = lane & 15;
  const int h  = lane >> 4;
  const int n  = lane & 15;

  int rowA = r0 + m;
  if (rowA >= N) rowA = N - 1;
  const float* aPtr  = agg1 + (size_t)rowA * NHID + 2 * h;
  const float* bBase = W2   + (size_t)(2 * h) * NCLASS + n;

  v8f acc = {};
#pragma unroll 4
  for (int k = 0; k < NHID; k += 4) {
    v2f a;
    a.x = fmaxf(aPtr[k + 0], 0.0f);           // ReLU on the fly
    a.y = fmaxf(aPtr[k + 1], 0.0f);
    v2f b;
    b.x = bBase[(size_t)k * NCLASS];
    b.y = bBase[(size_t)k * NCLASS + NCLASS];
    acc = __builtin_amdgcn_wmma_f32_16x16x4_f32(false, a, false, b, (short)0, acc, false, false);
  }

#pragma unroll
  for (int v = 0; v < 8; ++v) {
    const int row = r0 + v + 8 * h;
    if (row < N) g[(size_t)row * NCLASS + n] = acc[v];
  }
}

// ---------------------------------------------------------------------------
// Edge scatter, layer 1: agg1[dst] += w * h_pre[src]  (64 feats, 4 per thread)
// ---------------------------------------------------------------------------
__global__ void gcn_scatter64(const float* __restrict__ h,
                              const int* __restrict__ ei,
                              const float* __restrict__ ew,
                              float* __restrict__ agg, int E) {
  const long long idx = (long long)blockIdx.x * blockDim.x + threadIdx.x;
  const int e = (int)(idx >> 4);
  if (e >= E) return;
  const int f = ((int)idx & 15) * 4;
  const int s = ei[e];
  const int d = ei[E + e];
  const float w = ew[e];
  const float4 hv = *(const float4*)(h + (size_t)s * NHID + f);
  float* o = agg + (size_t)d * NHID + f;
  atomicAdd(o + 0, hv.x * w);
  atomicAdd(o + 1, hv.y * w);
  atomicAdd(o + 2, hv.z * w);
  atomicAdd(o + 3, hv.w * w);
}

// ---------------------------------------------------------------------------
// Edge scatter, layer 2: out[dst] += w * g[src]  (16 feats, 4 per thread)
// ---------------------------------------------------------------------------
__global__ void gcn_scatter16(const float* __restrict__ g,
                              const int* __restrict__ ei,
                              const float* __restrict__ ew,
                              float* __restrict__ out, int E) {
  const long long idx = (long long)blockIdx.x * blockDim.x + threadIdx.x;
  const int e = (int)(idx >> 2);
  if (e >= E) return;
  const int f = ((int)idx & 3) * 4;
  const int s = ei[e];
  const int d = ei[E + e];
  const float w = ew[e];
  const float4 gv = *(const float4*)(g + (size_t)s * NCLASS + f);
  float* o = out + (size_t)d * NCLASS + f;
  atomicAdd(o + 0, gv.x * w);
  atomicAdd(o + 1, gv.y * w);
  atomicAdd(o + 2, gv.z * w);
  atomicAdd(o + 3, gv.w * w);
}

// ---------------------------------------------------------------------------
extern "C" void kernel_launch(void* const* d_in, const int* in_sizes, int n_in,
                              void* d_out, int out_size, void* d_ws, size_t ws_size,
                              hipStream_t stream) {
  const float* x   = (const float*)d_in[0];
  const int*   ei1 = (const int*)  d_in[1];
  const int*   ei2 = (const int*)  d_in[2];
  const float* ew1 = (const float*)d_in[3];
  const float* ew2 = (const float*)d_in[4];
  const float* W1  = (const float*)d_in[5];
  const float* W2  = (const float*)d_in[6];
  float* out = (float*)d_out;

  const int N  = in_sizes[0] / NFEAT;   // 100000
  const int E1 = in_sizes[3];           // 1600000
  const int E2 = in_sizes[4];

  // workspace layout: h_pre [N,64] | agg1 [N,64] | g [N,16]  (~57.6 MB)
  float* h_pre = (float*)d_ws;
  float* agg1  = h_pre + (size_t)N * NHID;
  float* g     = agg1  + (size_t)N * NHID;

  // accumulators must start at zero every call (atomically accumulated)
  hipMemsetAsync(agg1, 0, (size_t)N * NHID   * sizeof(float), stream);
  hipMemsetAsync(out,  0, (size_t)N * NCLASS * sizeof(float), stream);

  const int nTiles     = (N + 15) / 16;        // 6250
  const int gemmBlocks = (nTiles + 7) / 8;     // 8 waves / block

  gcn_gemm1<<<gemmBlocks, 256, 0, stream>>>(x, W1, h_pre, nTiles, N);

  const long long t1 = (long long)E1 * 16;     // 4 feats per thread
  gcn_scatter64<<<(unsigned)((t1 + 255) / 256), 256, 0, stream>>>(h_pre, ei1, ew1, agg1, E1);

  gcn_gemm2<<<gemmBlocks, 256, 0, stream>>>(agg1, W2, g, nTiles, N);

  const long long t2 = (long long)E2 * 4;      // 4 feats per thread
  gcn_scatter16<<<(unsigned)((t2 + 255) / 256), 256, 0, stream>>>(g, ei2, ew2, out, E2);
}